// SiameseBrainNet_50371376447813
// MI455X (gfx1250) — compile-verified
//
#include <hip/hip_runtime.h>
#include <hip/hip_bf16.h>

// ---------------------------------------------------------------------------
// Siamese GCN encoder for MI455X (gfx1250, wave32).
// fp32 GEMM via V_WMMA_F32_16X16X4_F32; W-tiles double-buffered in LDS via
// GLOBAL_LOAD_ASYNC_TO_LDS_B128 (ASYNCcnt); edge aggregation via f32 atomics.
// ---------------------------------------------------------------------------

typedef float v2f __attribute__((ext_vector_type(2)));
typedef float v8f __attribute__((ext_vector_type(8)));

#define AS1 __attribute__((address_space(1)))
#define AS3 __attribute__((address_space(3)))

// Exact pointee type from the builtin's signature (clang diagnostic):
// '__attribute__((__vector_size__(4 * sizeof(int)))) int __device__ *'
typedef int b128_t __attribute__((__vector_size__(4 * sizeof(int))));

#if __has_builtin(__builtin_amdgcn_global_load_async_to_lds_b128)
#define HAVE_ASYNC_LDS 1
#else
#define HAVE_ASYNC_LDS 0
#endif

__device__ __forceinline__ void wait_async_lds()
{
#if HAVE_ASYNC_LDS
#if __has_builtin(__builtin_amdgcn_s_wait_asynccnt)
    __builtin_amdgcn_s_wait_asynccnt(0);
#else
    asm volatile("s_wait_asynccnt 0x0" ::: "memory");
#endif
#endif
}

#define TILE_M 128   // 8 waves x 16 rows
#define TILE_N 128   // 8 col-tiles of 16
#define TILE_K 32

// Stage W[kc:kc+TILE_K, colBase:colBase+TILE_N] into Ws (async if available).
__device__ __forceinline__ void stage_w(
    float (*Ws)[TILE_N], const float* __restrict__ W,
    int kc, int colBase, int Nc, int tid)
{
#pragma unroll
    for (int j = 0; j < (TILE_K * TILE_N) / 4 / 256; ++j) {
        int idx = tid + j * 256;
        int row = idx >> 5;                  // TILE_N/4 = 32 float4 per row
        int c4  = (idx & 31) << 2;
        const float* gp = W + (size_t)(kc + row) * Nc + colBase + c4;
        float*       lp = &Ws[row][c4];
#if HAVE_ASYNC_LDS
        __builtin_amdgcn_global_load_async_to_lds_b128(
            (AS1 b128_t*)(AS1 void*)(void*)const_cast<float*>(gp),
            (AS3 b128_t*)(AS3 void*)lp, 0, 0);
#else
        *(float4*)lp = *(const float4*)gp;
#endif
    }
}

// C[M,Nc] = A[M,K] * W[K,Nc]   (K % 32 == 0, Nc % 128 == 0)
__global__ __launch_bounds__(256) void gemm_f32_wmma(
    const float* __restrict__ A, const float* __restrict__ W,
    float* __restrict__ C, int M, int K, int Nc)
{
    __shared__ float Ws[2][TILE_K][TILE_N];   // 32 KB, double buffered

    const int tid     = threadIdx.x;
    const int wave    = tid >> 5;
    const int lane    = tid & 31;
    const int halfsel = lane >> 4;   // 0: lanes 0-15, 1: lanes 16-31
    const int lmod    = lane & 15;

    const int rowBase = blockIdx.x * TILE_M + wave * 16;
    const int colBase = blockIdx.y * TILE_N;

    v8f acc[TILE_N / 16];
#pragma unroll
    for (int t = 0; t < TILE_N / 16; ++t) acc[t] = (v8f)0.0f;

    // Clamp the A row: rows >= M compute garbage that is never stored.
    int aRow = rowBase + lmod;
    if (aRow > M - 1) aRow = M - 1;
    const float* aBase = A + (size_t)aRow * K;

    // Prologue: stage first chunk, wait for arrival.
    stage_w(Ws[0], W, 0, colBase, Nc, tid);
    wait_async_lds();
    __syncthreads();

    int buf = 0;
    for (int kc = 0; kc < K; kc += TILE_K) {
        // Issue async staging of the next chunk into the other buffer.
        if (kc + TILE_K < K) {
            stage_w(Ws[buf ^ 1], W, kc + TILE_K, colBase, Nc, tid);
            __builtin_prefetch(aBase + kc + TILE_K, 0, 1);
        }

#pragma unroll
        for (int ks = 0; ks < TILE_K; ks += 4) {
            // A fragment (16x4 f32): lane holds row lmod, k = ks + v + 2*halfsel
            const float* ap = aBase + kc + ks + 2 * halfsel;
            v2f afrag;
            afrag.x = ap[0];
            afrag.y = ap[1];
#pragma unroll
            for (int t = 0; t < TILE_N / 16; ++t) {
                // B fragment (4x16 f32): lane holds col lmod, k = ks + v + 2*halfsel
                v2f bfrag;
                bfrag.x = Ws[buf][ks + 2 * halfsel + 0][t * 16 + lmod];
                bfrag.y = Ws[buf][ks + 2 * halfsel + 1][t * 16 + lmod];
                acc[t] = __builtin_amdgcn_wmma_f32_16x16x4_f32(
                    /*neg_a=*/false, afrag, /*neg_b=*/false, bfrag,
                    /*c_mod=*/(short)0, acc[t],
                    /*reuse_a=*/false, /*reuse_b=*/false);
            }
        }

        // Next buffer fully landed (own asynccnt==0 for every wave) + all
        // waves done reading the current buffer before it is overwritten.
        wait_async_lds();
        __syncthreads();
        buf ^= 1;
    }

    // Store: acc[t][v] -> C[rowBase + v + 8*halfsel, colBase + t*16 + lmod]
    if (rowBase + 16 <= M) {
#pragma unroll
        for (int t = 0; t < TILE_N / 16; ++t) {
#pragma unroll
            for (int v = 0; v < 8; ++v) {
                int r = rowBase + v + 8 * halfsel;
                C[(size_t)r * Nc + colBase + t * 16 + lmod] = acc[t][v];
            }
        }
    } else {
#pragma unroll
        for (int t = 0; t < TILE_N / 16; ++t) {
#pragma unroll
            for (int v = 0; v < 8; ++v) {
                int r = rowBase + v + 8 * halfsel;
                if (r < M) {
                    C[(size_t)r * Nc + colBase + t * 16 + lmod] = acc[t][v];
                }
            }
        }
    }
}

// deg[dst[e]] += 1
__global__ void degree_k(const int* __restrict__ dst, float* __restrict__ deg, int E)
{
    int e = blockIdx.x * blockDim.x + threadIdx.x;
    if (e < E) atomicAdd(&deg[dst[e]], 1.0f);
}

// dinv[i] = rsqrt(deg[i] + 1)   (in place over the degree buffer)
__global__ void dinv_k(float* __restrict__ deg, int Nn)
{
    int i = blockIdx.x * blockDim.x + threadIdx.x;
    if (i < Nn) deg[i] = rsqrtf(deg[i] + 1.0f);
}

// One wave per edge: agg[dst] += h[src] * (dinv[src]*dinv[dst])
__global__ __launch_bounds__(256) void scatter_k(
    const float* __restrict__ h, const int* __restrict__ src,
    const int* __restrict__ dst, const float* __restrict__ dinv,
    float* __restrict__ agg, int E, int H)
{
    int gwave = (blockIdx.x * blockDim.x + threadIdx.x) >> 5;
    int lane  = threadIdx.x & 31;
    if (gwave >= E) return;
    int s = src[gwave];
    int d = dst[gwave];
    float w = dinv[s] * dinv[d];
    const float* hp = h + (size_t)s * H;
    float*       ap = agg + (size_t)d * H;
    for (int f = lane; f < H; f += 32) {
        atomicAdd(&ap[f], hp[f] * w);
    }
}

// One block per node: agg[n] = [relu](agg[n] + hw[n]*dinv[n]^2 + bias)
__global__ __launch_bounds__(256) void combine_k(
    float* __restrict__ agg, const float* __restrict__ hw,
    const float* __restrict__ dinv, const float* __restrict__ bias,
    int H, int relu)
{
    int node = blockIdx.x;
    float di = dinv[node];
    float di2 = di * di;
    size_t base = (size_t)node * H;
    for (int f = threadIdx.x; f < H; f += blockDim.x) {
        float v = agg[base + f] + hw[base + f] * di2 + bias[f];
        if (relu) v = fmaxf(v, 0.0f);
        agg[base + f] = v;
    }
}

// One wave per node: pooled[batch[n]] += h[n]; cnt[batch[n]] += 1
__global__ __launch_bounds__(256) void pool_k(
    const float* __restrict__ h, const int* __restrict__ batch,
    float* __restrict__ pooled, float* __restrict__ cnt, int Nn, int H)
{
    int gwave = (blockIdx.x * blockDim.x + threadIdx.x) >> 5;
    int lane  = threadIdx.x & 31;
    if (gwave >= Nn) return;
    int g = batch[gwave];
    if (lane == 0) atomicAdd(&cnt[g], 1.0f);
    const float* hp = h + (size_t)gwave * H;
    float*       pp = pooled + (size_t)g * H;
    for (int f = lane; f < H; f += 32) {
        atomicAdd(&pp[f], hp[f]);
    }
}

// out[g] = || pa[g]/ca[g] - pb[g]/cb[g] + 1e-6 ||_2   (one block per graph)
__global__ __launch_bounds__(256) void dist_k(
    const float* __restrict__ pa, const float* __restrict__ ca,
    const float* __restrict__ pb, const float* __restrict__ cb,
    float* __restrict__ out, int H)
{
    __shared__ float red[8];
    int g = blockIdx.x;
    float inva = 1.0f / fmaxf(ca[g], 1.0f);
    float invb = 1.0f / fmaxf(cb[g], 1.0f);
    float s = 0.0f;
    for (int f = threadIdx.x; f < H; f += blockDim.x) {
        float d = pa[(size_t)g * H + f] * inva - pb[(size_t)g * H + f] * invb + 1e-6f;
        s += d * d;
    }
#pragma unroll
    for (int off = 16; off > 0; off >>= 1) s += __shfl_down(s, off, 32);
    int lane = threadIdx.x & 31;
    int wv   = threadIdx.x >> 5;
    if (lane == 0) red[wv] = s;
    __syncthreads();
    if (threadIdx.x == 0) {
        float t = 0.0f;
        for (int i = 0; i < 8; ++i) t += red[i];
        out[g] = sqrtf(t);
    }
}

// ---------------------------------------------------------------------------

static void encode(const float* x, const int* ei, const int* batch,
                   const float* W1, const float* b1,
                   const float* W2, const float* b2,
                   const float* W3, const float* b3,
                   float* bufT, float* bufAgg, float* dinv,
                   float* pooled, float* cnt,
                   int Nn, int E, int D, int H, int G, hipStream_t stream)
{
    const int* src = ei;
    const int* dst = ei + E;
    size_t featBytes = (size_t)Nn * H * sizeof(float);

    // degree / normalization
    (void)hipMemsetAsync(dinv, 0, (size_t)Nn * sizeof(float), stream);
    degree_k<<<(E + 255) / 256, 256, 0, stream>>>(dst, dinv, E);
    dinv_k<<<(Nn + 255) / 256, 256, 0, stream>>>(dinv, Nn);

    dim3 ggrid((Nn + TILE_M - 1) / TILE_M, H / TILE_N);
    dim3 sgrid(((size_t)E * 32 + 255) / 256);

    // ---- layer 1: h1 = relu(agg(x@W1) + b1) ----
    gemm_f32_wmma<<<ggrid, 256, 0, stream>>>(x, W1, bufT, Nn, D, H);
    (void)hipMemsetAsync(bufAgg, 0, featBytes, stream);
    scatter_k<<<sgrid, 256, 0, stream>>>(bufT, src, dst, dinv, bufAgg, E, H);
    combine_k<<<Nn, 256, 0, stream>>>(bufAgg, bufT, dinv, b1, H, 1);

    // ---- layer 2 ----
    gemm_f32_wmma<<<ggrid, 256, 0, stream>>>(bufAgg, W2, bufT, Nn, H, H);
    (void)hipMemsetAsync(bufAgg, 0, featBytes, stream);
    scatter_k<<<sgrid, 256, 0, stream>>>(bufT, src, dst, dinv, bufAgg, E, H);
    combine_k<<<Nn, 256, 0, stream>>>(bufAgg, bufT, dinv, b2, H, 1);

    // ---- layer 3 (no relu) ----
    gemm_f32_wmma<<<ggrid, 256, 0, stream>>>(bufAgg, W3, bufT, Nn, H, H);
    (void)hipMemsetAsync(bufAgg, 0, featBytes, stream);
    scatter_k<<<sgrid, 256, 0, stream>>>(bufT, src, dst, dinv, bufAgg, E, H);
    combine_k<<<Nn, 256, 0, stream>>>(bufAgg, bufT, dinv, b3, H, 0);

    // ---- global mean pool (sums + counts; division fused into dist_k) ----
    (void)hipMemsetAsync(pooled, 0, (size_t)G * H * sizeof(float), stream);
    (void)hipMemsetAsync(cnt, 0, (size_t)G * sizeof(float), stream);
    pool_k<<<((size_t)Nn * 32 + 255) / 256, 256, 0, stream>>>(bufAgg, batch, pooled, cnt, Nn, H);
}

extern "C" void kernel_launch(void* const* d_in, const int* in_sizes, int n_in,
                              void* d_out, int out_size, void* d_ws, size_t ws_size,
                              hipStream_t stream)
{
    const float* x_a     = (const float*)d_in[0];
    const int*   ei_a    = (const int*)d_in[1];
    const int*   batch_a = (const int*)d_in[2];
    const float* x_b     = (const float*)d_in[3];
    const int*   ei_b    = (const int*)d_in[4];
    const int*   batch_b = (const int*)d_in[5];
    const float* W1      = (const float*)d_in[6];
    const float* b1      = (const float*)d_in[7];
    const float* W2      = (const float*)d_in[8];
    const float* b2      = (const float*)d_in[9];
    const float* W3      = (const float*)d_in[10];
    const float* b3      = (const float*)d_in[11];

    const int H  = in_sizes[7];             // 512
    const int D  = in_sizes[6] / H;         // 512
    const int Nn = in_sizes[0] / D;         // 20000
    const int E  = in_sizes[1] / 2;         // 640000
    const int G  = out_size;                // 64

    // workspace carve-up (256B aligned)
    char*  ws  = (char*)d_ws;
    size_t off = 0;
    auto carve = [&](size_t bytes) -> char* {
        char* p = ws + off;
        off += (bytes + 255) & ~((size_t)255);
        return p;
    };
    float* bufT     = (float*)carve((size_t)Nn * H * sizeof(float));
    float* bufAgg   = (float*)carve((size_t)Nn * H * sizeof(float));
    float* dinv     = (float*)carve((size_t)Nn * sizeof(float));
    float* pooled_a = (float*)carve((size_t)G * H * sizeof(float));
    float* pooled_b = (float*)carve((size_t)G * H * sizeof(float));
    float* cnt_a    = (float*)carve((size_t)G * sizeof(float));
    float* cnt_b    = (float*)carve((size_t)G * sizeof(float));
    (void)ws_size; (void)n_in;

    encode(x_a, ei_a, batch_a, W1, b1, W2, b2, W3, b3,
           bufT, bufAgg, dinv, pooled_a, cnt_a, Nn, E, D, H, G, stream);
    encode(x_b, ei_b, batch_b, W1, b1, W2, b2, W3, b3,
           bufT, bufAgg, dinv, pooled_b, cnt_b, Nn, E, D, H, G, stream);

    dist_k<<<G, 256, 0, stream>>>(pooled_a, cnt_a, pooled_b, cnt_b, (float*)d_out, H);
}